// ResidueSymmetryResolution_2370821947568
// MI455X (gfx1250) — compile-verified
//
#include <hip/hip_runtime.h>
#include <hip/hip_bf16.h>

// Problem constants (from reference)
#define BB 16
#define LL 2048
#define GG 8
#define PP 8
#define KK 64
#define CLAMP_V 15.0f
#define BIG_F 3.0e38f

typedef float v2f __attribute__((ext_vector_type(2)));
typedef float v8f __attribute__((ext_vector_type(8)));

// D = A(16x4 f32) x B(4x16 f32) + C(16x16 f32)  -> squared-distance tile
__device__ __forceinline__ v8f wmma_sqdist(v2f a, v2f b, v8f c) {
    return __builtin_amdgcn_wmma_f32_16x16x4_f32(
        /*neg_a=*/false, a, /*neg_b=*/false, b,
        /*c_mod=*/(short)0, c, /*reuse_a=*/false, /*reuse_b=*/false);
}

// single-instruction v_sqrt_f32 (~1 ulp; plenty for a scoring argmin)
__device__ __forceinline__ float fsqrt(float x) {
    return __builtin_amdgcn_sqrtf(x);
}

// v_med3_f32: one-instruction clamp; NaN input -> min of the other two
__device__ __forceinline__ float med3(float x, float lo, float hi) {
    return __builtin_amdgcn_fmed3f(x, lo, hi);
}

// ---------------------------------------------------------------------------
// init: copy x_native -> state, mask(bool) -> float state, zero all drms accums
// ---------------------------------------------------------------------------
__global__ __launch_bounds__(256) void init_kernel(
    const float* __restrict__ x_native_in,
    const unsigned char* __restrict__ mask_in,
    float* __restrict__ x_nat,
    float* __restrict__ nat_mask,
    float* __restrict__ drms_all)
{
    int i = blockIdx.x * blockDim.x + threadIdx.x;
    if (i < BB * LL * 3) x_nat[i] = x_native_in[i];
    if (i < BB * LL)     nat_mask[i] = mask_in[i] ? 1.0f : 0.0f;
    if (i < GG * BB * PP) drms_all[i] = 0.0f;
}

// ---------------------------------------------------------------------------
// drms: one wave handles one (batch b, 16-wide column tile of L).
// Accumulates sum_{k,l} min(nan0((dp-dn)^2),15)*colmask[l] into drms[b][p].
// ---------------------------------------------------------------------------
__global__ __launch_bounds__(256) void drms_kernel(
    const float* __restrict__ x_pred,
    const float* __restrict__ x_nat,
    const int*   __restrict__ amg,    // automorphs for this group: P*K ints
    float*       __restrict__ drms)   // B*P accumulators (pre-zeroed)
{
    __shared__ int sa[PP * KK];
    for (int i = threadIdx.x; i < PP * KK; i += blockDim.x) sa[i] = amg[i];
    __syncthreads();

    const int lane  = threadIdx.x & 31;
    const int wave  = blockIdx.x * (blockDim.x >> 5) + (threadIdx.x >> 5);
    const int LT    = LL / 16;            // 128 column tiles
    const int b     = wave / LT;
    const int ltile = wave % LT;
    const int n     = lane & 15;
    const bool hi   = lane >= 16;
    const int c     = ltile * 16 + n;     // global column index

    // column mask: 0 if this column is one of the base (a0) indices
    float cm = 1.0f;
    #pragma unroll 1
    for (int kk = 0; kk < KK; ++kk) if (sa[kk] == c) cm = 0.0f;

    // ---- column operands (fixed per wave) ----
    const float* yp = x_pred + (b * LL + c) * 3;
    float p0 = yp[0], p1 = yp[1], p2 = yp[2];
    float psq = p0 * p0 + p1 * p1 + p2 * p2;
    v2f Bp; Bp[0] = hi ? (-2.0f * p2) : (-2.0f * p0);
            Bp[1] = hi ? 1.0f         : (-2.0f * p1);
    v8f Cp;
    #pragma unroll
    for (int v = 0; v < 8; ++v) Cp[v] = psq;

    const float* yn = x_nat + (b * LL + c) * 3;
    float q0 = yn[0], q1 = yn[1], q2 = yn[2];
    float qsq = q0 * q0 + q1 * q1 + q2 * q2;
    v2f Bn; Bn[0] = hi ? (-2.0f * q2) : (-2.0f * q0);
            Bn[1] = hi ? 1.0f         : (-2.0f * q1);
    v8f Cn;
    #pragma unroll
    for (int v = 0; v < 8; ++v) Cn[v] = qsq;

    float acc[PP];
    #pragma unroll
    for (int p = 0; p < PP; ++p) acc[p] = 0.0f;

    #pragma unroll 1
    for (int kt = 0; kt < KK / 16; ++kt) {
        const int krow = kt * 16 + n;     // row within K (A-matrix M = lane&15)

        // ---- batch all index loads (LDS) ----
        const int rp = sa[krow];
        int rn[PP];
        #pragma unroll
        for (int p = 0; p < PP; ++p) rn[p] = sa[p * KK + krow];

        // ---- batch all coordinate gathers (global, one wait) ----
        const float* xp = x_pred + (b * LL + rp) * 3;
        float a0 = xp[0], a1 = xp[1], a2 = xp[2];
        float zc0[PP], zc1[PP], zc2[PP];
        #pragma unroll
        for (int p = 0; p < PP; ++p) {
            const float* xn = x_nat + (b * LL + rn[p]) * 3;
            zc0[p] = xn[0]; zc1[p] = xn[1]; zc2[p] = xn[2];
        }

        // ---- pred tile ----
        v2f Ap; Ap[0] = hi ? a2 : a0;
                Ap[1] = hi ? (a0 * a0 + a1 * a1 + a2 * a2) : a1;
        v8f Dp = wmma_sqdist(Ap, Bp, Cp);
        float dp[8];
        #pragma unroll
        for (int v = 0; v < 8; ++v) dp[v] = fsqrt(med3(Dp[v], 0.0f, BIG_F));

        // ---- nat tiles, fully unrolled: 8 pipelined WMMAs ----
        #pragma unroll
        for (int p = 0; p < PP; ++p) {
            float z0 = zc0[p], z1 = zc1[p], z2 = zc2[p];
            v2f An; An[0] = hi ? z2 : z0;
                    An[1] = hi ? (z0 * z0 + z1 * z1 + z2 * z2) : z1;
            v8f Dn = wmma_sqdist(An, Bn, Cn);

            float t = 0.0f;
            #pragma unroll
            for (int v = 0; v < 8; ++v) {
                float dn = fsqrt(med3(Dn[v], 0.0f, BIG_F));
                float df = dp[v] - dn;
                // min(nan_to_num(df^2), 15) in ONE v_med3_f32:
                // df^2 >= 0 -> clamp to [0,15]; NaN -> min(0,15)=0; inf -> 15
                t += med3(df * df, 0.0f, CLAMP_V);
            }
            acc[p] += t;
        }
    }

    // wave32 reduction + one atomic per (wave, p)
    #pragma unroll 1
    for (int p = 0; p < PP; ++p) {
        float r = acc[p] * cm;
        #pragma unroll
        for (int off = 16; off > 0; off >>= 1) r += __shfl_xor(r, off, 32);
        if (lane == 0) atomicAdd(drms + b * PP + p, r);
    }
}

// ---------------------------------------------------------------------------
// select: single block; argmin over p per batch, gather winning rows, sync,
// scatter into x_nat[:, a0] (matches reference's gather-before-update copy()).
// ---------------------------------------------------------------------------
__global__ __launch_bounds__(1024) void select_kernel(
    const int*   __restrict__ amg,     // P*K ints for this group
    const float* __restrict__ drms,    // B*P scores
    float*       __restrict__ x_nat,
    float*       __restrict__ nat_mask)
{
    const int tid = threadIdx.x;       // 0..1023 == B*K
    const int b = tid >> 6;            // / K
    const int k = tid & (KK - 1);

    // first-min argmin over the 8 permutations
    float best = drms[b * PP];
    int bestj = 0;
    #pragma unroll
    for (int j = 1; j < PP; ++j) {
        float v = drms[b * PP + j];
        if (v < best) { best = v; bestj = j; }
    }

    const int rsel = amg[bestj * KK + k];
    const float s0 = x_nat[(b * LL + rsel) * 3 + 0];
    const float s1 = x_nat[(b * LL + rsel) * 3 + 1];
    const float s2 = x_nat[(b * LL + rsel) * 3 + 2];
    const float sm = nat_mask[b * LL + rsel];

    __syncthreads();   // all gathers complete before any scatter

    const int wdst = amg[k];           // a0[k]
    x_nat[(b * LL + wdst) * 3 + 0] = s0;
    x_nat[(b * LL + wdst) * 3 + 1] = s1;
    x_nat[(b * LL + wdst) * 3 + 2] = s2;
    nat_mask[b * LL + wdst] = sm;
}

// ---------------------------------------------------------------------------
extern "C" void kernel_launch(void* const* d_in, const int* in_sizes, int n_in,
                              void* d_out, int out_size, void* d_ws, size_t ws_size,
                              hipStream_t stream) {
    const float*         x_pred   = (const float*)d_in[0];
    const float*         x_native = (const float*)d_in[1];
    const unsigned char* x_mask   = (const unsigned char*)d_in[2]; // bool (1B)
    const int*           autom    = (const int*)d_in[3];           // (G,P,K) int32

    float* x_nat    = (float*)d_out;              // B*L*3 floats
    float* nat_mask = x_nat + BB * LL * 3;        // B*L floats (0/1)
    float* drms_all = (float*)d_ws;               // G*B*P floats

    // init state + zero score accumulators (deterministic every call)
    {
        int n = BB * LL * 3;
        init_kernel<<<(n + 255) / 256, 256, 0, stream>>>(
            x_native, x_mask, x_nat, nat_mask, drms_all);
    }

    // sequential scan over the 8 automorphism groups
    const int wavesTotal = BB * (LL / 16);        // 2048 waves
    const int blocks = wavesTotal / 8;            // 8 waves / 256-thread block
    for (int g = 0; g < GG; ++g) {
        const int*   amg = autom + g * PP * KK;
        float*       drm = drms_all + g * BB * PP;
        drms_kernel<<<blocks, 256, 0, stream>>>(x_pred, x_nat, amg, drm);
        select_kernel<<<1, BB * KK, 0, stream>>>(amg, drm, x_nat, nat_mask);
    }
}